// online_triplet_loss_24842090840333
// MI455X (gfx1250) — compile-verified
//
#include <hip/hip_runtime.h>
#include <math.h>

// ---------------------------------------------------------------------------
// Online triplet loss for MI455X (gfx1250, wave32).
// d[i,j] = q_i + q_j - 2*G[i,j] + 2*eps*(s_i - s_j) + D*eps^2,  G = f f^T
// G computed with V_WMMA_F32_16X16X4_F32 (fp32 matrix pipe, exact precision).
// All reductions are fixed-order LDS trees -> deterministic across replays.
// ---------------------------------------------------------------------------

typedef float v2f __attribute__((ext_vector_type(2)));
typedef float v8f __attribute__((ext_vector_type(8)));

constexpr int   N    = 512;
constexpr int   D    = 256;
constexpr float EPS  = 1e-6f;

// ---------------------------------------------------------------------------
// Kernel 1: row L2-normalize; also emit s_i = sum(f_i), q_i = sum(f_i^2)
// One wave32 per row; 8 waves per 256-thread block.
// ---------------------------------------------------------------------------
__global__ __launch_bounds__(256) void k_normalize(const float* __restrict__ x,
                                                   float* __restrict__ f,
                                                   float* __restrict__ s,
                                                   float* __restrict__ q) {
    const int wave = threadIdx.x >> 5;
    const int lane = threadIdx.x & 31;
    const int row  = blockIdx.x * 8 + wave;

    const float* xr = x + row * D;
    float v[8];
    float ss = 0.0f;
#pragma unroll
    for (int t = 0; t < 8; ++t) {            // coalesced: lane + 32*t
        v[t] = xr[lane + 32 * t];
        ss  += v[t] * v[t];
    }
#pragma unroll
    for (int m = 16; m >= 1; m >>= 1) ss += __shfl_xor(ss, m, 32);

    const float inv = 1.0f / sqrtf(ss);      // matches f = x / ||x||
    float* fr = f + row * D;
    float sn = 0.0f, qn = 0.0f;
#pragma unroll
    for (int t = 0; t < 8; ++t) {
        const float fv = v[t] * inv;
        fr[lane + 32 * t] = fv;
        sn += fv;
        qn += fv * fv;
    }
#pragma unroll
    for (int m = 16; m >= 1; m >>= 1) {
        sn += __shfl_xor(sn, m, 32);
        qn += __shfl_xor(qn, m, 32);
    }
    if (lane == 0) { s[row] = sn; q[row] = qn; }
}

// ---------------------------------------------------------------------------
// Kernel 2: G = f f^T via V_WMMA_F32_16X16X4_F32, fused distance epilogue.
// One wave per 16x16 tile; 8 waves (8 tiles) per block; 32x32 tile grid.
//
// A 16x4 f32 layout : lane L -> row (L&15), K = 2*(L>>4) + {0,1} in v2f.
// B 4x16 f32 layout : mirror of A; since B = f^T, per-lane B data is the same
//                     pattern with the column-tile row base -> float2 loads.
// C/D 16x16 layout  : VGPR r -> M = r + 8*(lane>>4), N = lane&15.
// ---------------------------------------------------------------------------
__global__ __launch_bounds__(256) void k_gram_dist(const float* __restrict__ f,
                                                   const float* __restrict__ s,
                                                   const float* __restrict__ q,
                                                   float* __restrict__ dmat) {
    const int wave = threadIdx.x >> 5;
    const int lane = threadIdx.x & 31;
    const int tile = blockIdx.x * 8 + wave;
    const int tI   = tile >> 5;     // row-tile index   (0..31)
    const int tJ   = tile & 31;     // col-tile index   (0..31)

    const int half = lane >> 4;     // 0: K lo pair, 1: K hi pair
    const int m    = lane & 15;

    const float* arow = f + (tI * 16 + m) * D + 2 * half;
    const float* brow = f + (tJ * 16 + m) * D + 2 * half;

    v8f acc = {0.f, 0.f, 0.f, 0.f, 0.f, 0.f, 0.f, 0.f};

#pragma unroll 8
    for (int k0 = 0; k0 < D; k0 += 4) {
        const v2f a = *(const v2f*)(arow + k0);   // 8B-aligned
        const v2f b = *(const v2f*)(brow + k0);
        acc = __builtin_amdgcn_wmma_f32_16x16x4_f32(
            /*neg_a=*/false, a, /*neg_b=*/false, b,
            /*c_mod=*/(short)0, acc, /*reuse_a=*/false, /*reuse_b=*/false);
    }

    // Epilogue: d[i,j] = q_i + q_j - 2G + 2*eps*(s_i - s_j) + D*eps^2
    const int   j  = tJ * 16 + m;               // N = lane&15
    const float sj = s[j];
    const float qj = q[j];
    const float c0 = (float)D * EPS * EPS;
#pragma unroll
    for (int r = 0; r < 8; ++r) {
        const int   i  = tI * 16 + r + 8 * half; // M = r + 8*half
        const float g  = acc[r];
        const float dv = q[i] + qj - 2.0f * g + 2.0f * EPS * (s[i] - sj) + c0;
        dmat[i * N + j] = dv;
    }
}

// ---------------------------------------------------------------------------
// Kernel 3: per-anchor triplet partials. One 256-thread block per anchor.
// Labels form contiguous sorted blocks; pos = [first,last) \ {i}, neg =
// (<first) U (>=last). Positives staged in LDS; threads stride negatives.
// ---------------------------------------------------------------------------
__global__ __launch_bounds__(256) void k_triplet(const float* __restrict__ dmat,
                                                 const int* __restrict__ labels,
                                                 const float* __restrict__ margin_p,
                                                 float* __restrict__ part_loss,
                                                 float* __restrict__ part_ge,
                                                 float* __restrict__ part_cnt) {
    __shared__ int      sh_first, sh_last;
    __shared__ float    sh_dpos[N];
    __shared__ float    red_f[256];
    __shared__ unsigned red_u[256];

    const int i = blockIdx.x;
    const int t = threadIdx.x;

    if (t == 0) { sh_first = N; sh_last = -1; }
    __syncthreads();

    const int lab = labels[i];
    int lmin = N, lmax = -1;
    for (int idx = t; idx < N; idx += 256) {
        if (labels[idx] == lab) {
            lmin = (idx < lmin) ? idx : lmin;
            lmax = (idx > lmax) ? idx : lmax;
        }
    }
    atomicMin(&sh_first, lmin);
    atomicMax(&sh_last,  lmax);
    __syncthreads();

    const int first = sh_first;
    const int last  = sh_last;
    const float* drow = dmat + i * N;

    // compacted positive distances -> LDS
    const int npos = (last - first) - ((i < last) ? 1 : 0);
    for (int idx = first + t; idx < last; idx += 256) {
        if (idx == i) continue;
        const int p = (idx - first) - ((i < last && idx > i) ? 1 : 0);
        sh_dpos[p] = drow[idx];
    }
    __syncthreads();

    const float margin = *margin_p;
    float    lsum = 0.0f;
    unsigned ge   = 0u;
    for (int k = t; k < N; k += 256) {
        if (k >= first && k < last) continue;    // keep negatives only
        const float dan = drow[k];
        for (int p = 0; p < npos; ++p) {
            const float diff = sh_dpos[p] - dan; // d_ap - d_an
            lsum += fmaxf(diff + margin, 0.0f);
            ge   += (diff >= 0.0f) ? 1u : 0u;
        }
    }

    red_f[t] = lsum;
    red_u[t] = ge;
    __syncthreads();
    for (int st = 128; st >= 1; st >>= 1) {
        if (t < st) { red_f[t] += red_f[t + st]; red_u[t] += red_u[t + st]; }
        __syncthreads();
    }
    if (t == 0) {
        const int nneg = first + (N - last);
        part_loss[i] = red_f[0];
        part_ge[i]   = (float)red_u[0];
        part_cnt[i]  = (float)(npos * nneg);
    }
}

// ---------------------------------------------------------------------------
// Kernel 4: deterministic final reduction over 512 per-anchor partials.
// ---------------------------------------------------------------------------
__global__ __launch_bounds__(256) void k_finalize(const float* __restrict__ part_loss,
                                                  const float* __restrict__ part_ge,
                                                  const float* __restrict__ part_cnt,
                                                  float* __restrict__ out) {
    __shared__ float rl[256], rg[256], rc[256];
    const int t = threadIdx.x;
    rl[t] = part_loss[t] + part_loss[t + 256];
    rg[t] = part_ge[t]   + part_ge[t + 256];
    rc[t] = part_cnt[t]  + part_cnt[t + 256];
    __syncthreads();
    for (int st = 128; st >= 1; st >>= 1) {
        if (t < st) { rl[t] += rl[t + st]; rg[t] += rg[t + st]; rc[t] += rc[t + st]; }
        __syncthreads();
    }
    if (t == 0) {
        const float cnt = rc[0];
        out[0] = rl[0] / cnt;            // loss
        out[1] = 1.0f - rg[0] / cnt;     // accuracy
    }
}

// ---------------------------------------------------------------------------
// Host-side launch. Workspace layout (all fp32, well within typical ws_size):
//   f[N*D] | s[N] | q[N] | dmat[N*N] | part_loss[N] | part_ge[N] | part_cnt[N]
//   = 512KB + 2KB + 2KB + 1MB + 3*2KB  ~= 1.55 MB
// ---------------------------------------------------------------------------
extern "C" void kernel_launch(void* const* d_in, const int* in_sizes, int n_in,
                              void* d_out, int out_size, void* d_ws, size_t ws_size,
                              hipStream_t stream) {
    const float* x      = (const float*)d_in[0];
    const int*   labels = (const int*)d_in[1];     // JAX x64 off -> int32
    const float* margin = (const float*)d_in[2];

    char*  ws        = (char*)d_ws;
    float* f         = (float*)ws;                           // N*D
    float* s         = (float*)(ws + (size_t)N * D * 4);     // N
    float* q         = s + N;                                // N
    float* dmat      = q + N;                                // N*N
    float* part_loss = dmat + (size_t)N * N;                 // N
    float* part_ge   = part_loss + N;                        // N
    float* part_cnt  = part_ge + N;                          // N

    float* out = (float*)d_out;

    k_normalize<<<N / 8, 256, 0, stream>>>(x, f, s, q);
    k_gram_dist<<<(N / 16) * (N / 16) / 8, 256, 0, stream>>>(f, s, q, dmat);
    k_triplet  <<<N, 256, 0, stream>>>(dmat, labels, margin,
                                       part_loss, part_ge, part_cnt);
    k_finalize <<<1, 256, 0, stream>>>(part_loss, part_ge, part_cnt, out);
}